// CapsuleNetwork_84490596647312
// MI455X (gfx1250) — compile-verified
//
#include <hip/hip_runtime.h>

typedef __attribute__((ext_vector_type(2))) float v2f;
typedef __attribute__((ext_vector_type(8))) float v8f;

#define BATCH   16384
#define IN_DIM  1024
#define OUT_DIM 512
#define N_CAPS  8
#define CAP_DIM 16
#define ND      128   // N_CAPS*CAP_DIM

// ---- kernel 1 tiling ----
#define K1_KC    64   // K-chunk staged in LDS
#define K1_NCHNK (IN_DIM / K1_KC)   // 16 chunks
#define K1_XP    68   // xs row pitch (floats): banks (4*row+k)%64 -> conflict-free
#define K1_WP2   288  // packed Wp pair-row pitch: 288%64==32 -> halves on disjoint banks
#define XS_BUF   (128 * K1_XP)        // floats per xs buffer
#define WP_BUF   ((K1_KC / 2) * K1_WP2)

// ---- kernel 2 LDS pitches ----
#define W_P2 1064     // packed W_flat pair-row pitch (1064%64==40)
#define V_P2 40       // packed v pair-row pitch
#define P_P  132      // primary tile pitch

__device__ __forceinline__ float squash_scale(float sq) {
    // (sq/(1+sq)) / sqrt(sq + 1e-8)
    return (sq / (1.0f + sq)) * __frsqrt_rn(sq + 1e-8f);
}

// CDNA5 async DMA: memory -> LDS, 16B per lane, tracked by ASYNCcnt.
__device__ __forceinline__ void async_load_b128(const float* gsrc, unsigned lds_byte_off) {
    asm volatile("global_load_async_to_lds_b128 %0, %1, off"
                 :: "v"(lds_byte_off), "v"(gsrc)
                 : "memory");
}
__device__ __forceinline__ void wait_async0() {
    asm volatile("s_wait_asynccnt 0x0" ::: "memory");
}

// =====================================================================
// Kernel 1: primary = squash(reshape(x @ Wp + bp))  -> d_ws [B][128]
// Grid: 128 blocks x 256 threads. Each wave: 16 rows x 128 cols.
// Double-buffered pipeline:
//   x tile   : global_load_async_to_lds_b128 (no VGPR round-trip)
//   Wp chunk : synchronous pair-interleave pack (overlaps compute, other buffer)
// Wp packed: element (k,n) at wp2[(k>>1)*288 + n*2 + (k&1)] -> B frag = one b64.
// =====================================================================
extern "C" __global__ void __launch_bounds__(256)
caps_primary_kernel(const float* __restrict__ x, const float* __restrict__ Wp,
                    const float* __restrict__ bp, float* __restrict__ primary) {
    extern __shared__ float lds[];
    float* xs  = lds;                  // [2][128][K1_XP]
    float* wp2 = lds + 2 * XS_BUF;     // [2][K1_KC/2][K1_WP2]
    const unsigned wp2_byte_base = 2 * XS_BUF * 4;
    (void)wp2_byte_base;

    const int tid  = threadIdx.x;
    const int wave = tid >> 5;
    const int lane = tid & 31;
    const int half = lane >> 4;
    const int l16  = lane & 15;
    const int r0   = blockIdx.x * 128;

    // ---- async x-tile issue: each thread DMAs 8 x 16B ----
    const int xc4 = tid & 15;     // float4 column
    const int xrb = tid >> 4;     // row base (16 rows per pass)
    auto issue_x_chunk = [&](int k0, int buf) {
#pragma unroll
        for (int p = 0; p < 8; ++p) {
            const int row = xrb + p * 16;
            async_load_b128(x + (size_t)(r0 + row) * IN_DIM + k0 + xc4 * 4,
                            (unsigned)((buf * XS_BUF + row * K1_XP + xc4 * 4) * 4));
        }
    };
    // ---- synchronous Wp chunk load + pair-interleave pack ----
    const int wc4 = tid & 31;     // float4 col (0..31 -> 128 floats)
    const int wpb = tid >> 5;     // pair-row base (8 per pass)
    auto load_wp_chunk = [&](int k0, int buf) {
#pragma unroll
        for (int p = 0; p < 4; ++p) {
            const int pr = wpb + p * 8;          // pair-row 0..31
            const float4 a4 = *(const float4*)(Wp + (size_t)(k0 + 2 * pr) * ND + wc4 * 4);
            const float4 b4 = *(const float4*)(Wp + (size_t)(k0 + 2 * pr + 1) * ND + wc4 * 4);
            float* dst = wp2 + buf * WP_BUF + pr * K1_WP2 + wc4 * 8;   // (c4*4)*2
            v2f p0; p0.x = a4.x; p0.y = b4.x; *(v2f*)(dst + 0) = p0;
            v2f p1; p1.x = a4.y; p1.y = b4.y; *(v2f*)(dst + 2) = p1;
            v2f p2; p2.x = a4.z; p2.y = b4.z; *(v2f*)(dst + 4) = p2;
            v2f p3; p3.x = a4.w; p3.y = b4.w; *(v2f*)(dst + 6) = p3;
        }
    };

    v8f acc[8] = {};

    // prologue: start chunk 0
    issue_x_chunk(0, 0);
    load_wp_chunk(0, 0);

    const int rowl = wave * 16 + l16;
    for (int i = 0; i < K1_NCHNK; ++i) {
        const int cb = i & 1;
        wait_async0();          // chunk i DMA complete (this wave's share)
        __syncthreads();        // all waves' shares + wp2[cb] stores visible
        if (i + 1 < K1_NCHNK) { // prefetch chunk i+1 into the other buffer
            issue_x_chunk((i + 1) * K1_KC, cb ^ 1);
            load_wp_chunk((i + 1) * K1_KC, cb ^ 1);
        }
        const float* xsb = xs + cb * XS_BUF;
        const float* wpb2 = wp2 + cb * WP_BUF;
        for (int k = 0; k < K1_KC; k += 4) {
            const int kk = k + 2 * half;
            // A fragment (16x4 f32): lanes 0-15 K=k,k+1 ; lanes 16-31 K=k+2,k+3
            v2f a = *(const v2f*)(xsb + rowl * K1_XP + kk);
            const float* brow = wpb2 + (kk >> 1) * K1_WP2 + l16 * 2;
#pragma unroll
            for (int t = 0; t < 8; ++t) {
                v2f b = *(const v2f*)(brow + t * 32);   // (t*16)*2
                acc[t] = __builtin_amdgcn_wmma_f32_16x16x4_f32(
                    false, a, false, b, (short)0, acc[t], false, false);
            }
        }
    }

    // bias + capsule squash (tile t == capsule t) + store
#pragma unroll
    for (int t = 0; t < 8; ++t) {
        const float bb = bp[t * 16 + l16];
#pragma unroll
        for (int j = 0; j < 8; ++j) acc[t][j] += bb;
#pragma unroll
        for (int j = 0; j < 8; ++j) {
            float val = acc[t][j];
            float sq = val * val;
#pragma unroll
            for (int off = 8; off >= 1; off >>= 1) sq += __shfl_xor(sq, off, 16);
            val *= squash_scale(sq);
            const int grow = r0 + wave * 16 + j + 8 * half;   // C layout: vgpr j -> M=j / j+8
            primary[(size_t)grow * ND + t * 16 + l16] = val;
        }
    }
}

// =====================================================================
// Kernel 2: dynamic routing, W_flat resident in LDS, pair-interleaved:
//   element (nd,o) at Wp2[(nd>>1)*W_P2 + o*2 + (nd&1)]
// s-GEMM B frags and t-GEMM (transposed form) B frags are single b64 loads.
// =====================================================================
extern "C" __global__ void __launch_bounds__(256)
caps_routing_kernel(const float* __restrict__ primary, const float* __restrict__ W,
                    const int* __restrict__ n_routing_p, float* __restrict__ out,
                    int tilesPerWG) {
    extern __shared__ float lds[];
    float* Wp2   = lds;                    // [64][W_P2]   packed W_flat
    float* vb2   = Wp2 + 64 * W_P2;        // [256][V_P2]  packed v: (row,o)->[(o>>1)*V_P2 + row*2 + (o&1)]
    float* ptile = vb2 + 256 * V_P2;       // [16][P_P]
    float* blog  = ptile + 16 * P_P;       // [16][8] routing logits
    float* crow  = blog + 128;             // [16][8] softmax coeffs
    float* ssq   = crow + 128;             // [16]    |s|^2 per row

    const int tid  = threadIdx.x;
    const int wave = tid >> 5;
    const int lane = tid & 31;
    const int half = lane >> 4;
    const int l16  = lane & 15;
    const int nr   = *n_routing_p;

    // Load W_flat [128][512] into packed LDS layout
    for (int i = 0; i < 32; ++i) {
        const int idx = i * 256 + tid;         // 8192 float4-pair tasks
        const int pr  = idx >> 7;              // nd pair-row 0..63
        const int c4  = idx & 127;             // float4 col
        const float4 a4 = *(const float4*)(W + (size_t)(2 * pr) * OUT_DIM + c4 * 4);
        const float4 b4 = *(const float4*)(W + (size_t)(2 * pr + 1) * OUT_DIM + c4 * 4);
        float* dst = Wp2 + pr * W_P2 + c4 * 8;
        v2f p0; p0.x = a4.x; p0.y = b4.x; *(v2f*)(dst + 0) = p0;
        v2f p1; p1.x = a4.y; p1.y = b4.y; *(v2f*)(dst + 2) = p1;
        v2f p2; p2.x = a4.z; p2.y = b4.z; *(v2f*)(dst + 4) = p2;
        v2f p3; p3.x = a4.w; p3.y = b4.w; *(v2f*)(dst + 6) = p3;
    }
    __syncthreads();

    for (int tt = 0; tt < tilesPerWG; ++tt) {
        const int tile = blockIdx.x * tilesPerWG + tt;
        const int r0 = tile * 16;

        // load primary tile 16x128
        {
            const int row = tid >> 4;
            const int c4  = tid & 15;
            const float4 v0 = *(const float4*)(primary + (size_t)(r0 + row) * ND + c4 * 4);
            const float4 v1 = *(const float4*)(primary + (size_t)(r0 + row) * ND + (c4 + 16) * 4);
            *(float4*)(ptile + row * P_P + c4 * 4) = v0;
            *(float4*)(ptile + row * P_P + (c4 + 16) * 4) = v1;
        }
        if (tid < 128) blog[tid] = 0.0f;
        __syncthreads();

        for (int it = 0; it < nr; ++it) {
            // softmax over 8 capsules per row; zero |s|^2 accumulators
            if (tid < 16) {
                float m = blog[tid * 8];
#pragma unroll
                for (int n = 1; n < 8; ++n) m = fmaxf(m, blog[tid * 8 + n]);
                float s = 0.0f, e[8];
#pragma unroll
                for (int n = 0; n < 8; ++n) { e[n] = __expf(blog[tid * 8 + n] - m); s += e[n]; }
                const float inv = 1.0f / s;
#pragma unroll
                for (int n = 0; n < 8; ++n) crow[tid * 8 + n] = e[n] * inv;
                ssq[tid] = 0.0f;
            }
            __syncthreads();

            // s-GEMM: s[16, wave*64 .. +64) = (c .* primary) @ W_flat
            // capsule-major k loop: hoist softmax coeff (1 DS read / 4 k-steps)
            v8f acc[4] = {};
            for (int cap = 0; cap < N_CAPS; ++cap) {
                const float cw = crow[l16 * 8 + cap];
#pragma unroll
                for (int k2 = 0; k2 < CAP_DIM; k2 += 4) {
                    const int kk = cap * CAP_DIM + k2 + 2 * half;
                    v2f a = *(const v2f*)(ptile + l16 * P_P + kk);
                    a.x *= cw; a.y *= cw;
                    const float* brow = Wp2 + (kk >> 1) * W_P2 + (wave * 64 + l16) * 2;
#pragma unroll
                    for (int t = 0; t < 4; ++t) {
                        v2f b = *(const v2f*)(brow + t * 32);   // (t*16)*2
                        acc[t] = __builtin_amdgcn_wmma_f32_16x16x4_f32(
                            false, a, false, b, (short)0, acc[t], false, false);
                    }
                }
            }
            // per-row partial |s|^2 -> LDS float atomics (ds_add_f32)
#pragma unroll
            for (int j = 0; j < 8; ++j) {
                float p = 0.0f;
#pragma unroll
                for (int t = 0; t < 4; ++t) { const float v = acc[t][j]; p += v * v; }
#pragma unroll
                for (int off = 8; off >= 1; off >>= 1) p += __shfl_xor(p, off, 16);
                if (l16 == 0) atomicAdd(&ssq[j + 8 * half], p);
            }
            __syncthreads();

            // v = squash(s): scale, store packed v (and d_out on last iter)
            const bool last = (it == nr - 1);
#pragma unroll
            for (int j = 0; j < 8; ++j) {
                const int row = j + 8 * half;
                const float sc = squash_scale(ssq[row]);
#pragma unroll
                for (int t = 0; t < 4; ++t) {
                    const int col = wave * 64 + t * 16 + l16;
                    const float v = acc[t][j] * sc;
                    vb2[(col >> 1) * V_P2 + row * 2 + (col & 1)] = v;
                    if (last) out[(size_t)(r0 + row) * OUT_DIM + col] = v;
                }
            }
            __syncthreads();

            if (!last) {
                // t-GEMM (transposed): t'[M=nd in cap chunk][N=batch row] = W-chunk x v
                // wave == capsule. A from packed W (nd rows), B from packed v (b64).
                v8f tacc = {};
                const int nd = wave * 16 + l16;
                const float* abase = Wp2 + (nd >> 1) * W_P2 + (nd & 1);
                for (int k = 0; k < OUT_DIM; k += 4) {
                    const int kk = k + 2 * half;
                    v2f a; a.x = abase[kk * 2]; a.y = abase[kk * 2 + 2];
                    v2f b = *(const v2f*)(vb2 + (kk >> 1) * V_P2 + l16 * 2);
                    tacc = __builtin_amdgcn_wmma_f32_16x16x4_f32(
                        false, a, false, b, (short)0, tacc, false, false);
                }
                // agreement[row=l16, cap=wave] = sum_d t'[d][row] * primary[row][cap,d]
                // lane holds d = j + 8*half for j=0..7 at batch row l16.
                float part = 0.0f;
#pragma unroll
                for (int j = 0; j < 8; ++j) {
                    part += tacc[j] * ptile[l16 * P_P + wave * 16 + j + 8 * half];
                }
                part += __shfl_xor(part, 16, 32);   // combine the two halves
                if (half == 0) blog[l16 * 8 + wave] += part;  // unique (row,cap) writer
            }
            __syncthreads();
        }
        __syncthreads();   // tile state (ptile/blog) safe to overwrite
    }
}

// =====================================================================
// Host launch
// =====================================================================
extern "C" void kernel_launch(void* const* d_in, const int* in_sizes, int n_in,
                              void* d_out, int out_size, void* d_ws, size_t ws_size,
                              hipStream_t stream) {
    const float* x  = (const float*)d_in[0];   // [16384,1024]
    const float* Wp = (const float*)d_in[1];   // [1024,128]
    const float* bp = (const float*)d_in[2];   // [128]
    const float* W  = (const float*)d_in[3];   // [8,16,512] == W_flat [128,512]
    const int*   nr = (const int*)d_in[4];     // n_routing (device scalar)
    float* out      = (float*)d_out;           // [16384,512]
    float* primary  = (float*)d_ws;            // [16384,128] scratch

    const int k1_lds = (2 * XS_BUF + 2 * WP_BUF) * 4;                            // 143,360 B
    const int k2_lds = (64 * W_P2 + 256 * V_P2 + 16 * P_P + 128 + 128 + 16) * 4; // 322,880 B

    (void)hipFuncSetAttribute((const void*)caps_primary_kernel,
                              hipFuncAttributeMaxDynamicSharedMemorySize, k1_lds);
    (void)hipFuncSetAttribute((const void*)caps_routing_kernel,
                              hipFuncAttributeMaxDynamicSharedMemorySize, k2_lds);

    caps_primary_kernel<<<BATCH / 128, 256, k1_lds, stream>>>(x, Wp, bp, primary);

    const int numTiles = BATCH / 16;         // 1024
    const int grid2 = 256;
    const int tilesPerWG = numTiles / grid2; // 4
    caps_routing_kernel<<<grid2, 256, k2_lds, stream>>>(primary, W, nr, out, tilesPerWG);
}